// Node_Layer_83949430768186
// MI455X (gfx1250) — compile-verified
//
#include <hip/hip_runtime.h>
#include <hip/hip_bf16.h>
#include <stddef.h>

typedef __attribute__((ext_vector_type(2))) float v2f;
typedef __attribute__((ext_vector_type(4))) float v4f;
typedef __attribute__((ext_vector_type(8))) float v8f;

#define NODE_NF 128
#define EDGE_NF 64
#define IN_NF   192
#define H_NF    256
#define OUT_NF  128

// LDS row strides padded by 4 floats: 196 % 64 == 4, 260 % 64 == 4 -> 4-bank
// skew per row, so the 16 distinct-row b64 A-fragment reads are conflict-free.
#define XSTR 196
#define HSTR 260

// ---------------------------------------------------------------------------
// Kernel 1: zero the aggregation scratch (N*64 floats) with b128 stores.
// ---------------------------------------------------------------------------
__global__ void zero_agg(float* __restrict__ agg, int n4) {
    int i = blockIdx.x * blockDim.x + threadIdx.x;
    if (i < n4) {
        v4f z = {0.f, 0.f, 0.f, 0.f};
        *(v4f*)(agg + (size_t)i * 4) = z;
    }
}

// ---------------------------------------------------------------------------
// Kernel 2: scatter-add edge_attr rows into agg[row[e]].
// 16 threads per edge; each thread: 1x global_load_b128 + 4x atomic_add_f32.
// agg (12.8 MB) stays resident in the 192 MB L2, so atomics complete at L2.
// ---------------------------------------------------------------------------
__global__ void scatter_edges(const float* __restrict__ edge_attr,
                              const int*   __restrict__ row,
                              float* __restrict__ agg, int E) {
    int t = blockIdx.x * blockDim.x + threadIdx.x;
    int e = t >> 4;
    if (e >= E) return;
    int c = (t & 15) << 2;
    int r = row[e];
    v4f v = *(const v4f*)(edge_attr + (size_t)e * EDGE_NF + c);
    float* dst = agg + (size_t)r * EDGE_NF + c;
    unsafeAtomicAdd(dst + 0, v.x);
    unsafeAtomicAdd(dst + 1, v.y);
    unsafeAtomicAdd(dst + 2, v.z);
    unsafeAtomicAdd(dst + 3, v.w);
}

// ---------------------------------------------------------------------------
// Kernel 3: repack a row-major weight matrix W[K x C] into WMMA B-fragment
// order: fragment g = (nt*(K/4) + kq), lane l (= half*16 + lr) holds
// {W[kq*4 + 2*half][nt*16+lr], W[kq*4 + 2*half + 1][nt*16+lr]} as one float2
// at WT[g*32 + l].  In the GEMM loop one wave-wide global_load_b64 then
// fetches a whole B fragment as one contiguous 256 B request.
// ---------------------------------------------------------------------------
__global__ void make_frags(const float* __restrict__ W, v2f* __restrict__ WT,
                           int K, int C) {
    int tid = blockIdx.x * blockDim.x + threadIdx.x;
    int total = (K / 4) * (C / 16) * 32;
    if (tid >= total) return;
    int lane = tid & 31;
    int g    = tid >> 5;
    int kq   = g % (K / 4);
    int nt   = g / (K / 4);
    int k    = kq * 4 + ((lane >> 4) << 1);
    int n    = nt * 16 + (lane & 15);
    v2f b;
    b.x = W[(size_t)k * C + n];
    b.y = W[(size_t)(k + 1) * C + n];
    WT[tid] = b;
}

// ---------------------------------------------------------------------------
// Kernel 4: fused concat + MLP via V_WMMA_F32_16X16X4_F32.
// One wave32 per 16-node tile. X tile staged into padded LDS with
// GLOBAL_LOAD_ASYNC_TO_LDS_B128 (ASYNCcnt path); h tile bounced through LDS
// to transpose the WMMA D layout (lane=column) into the A layout (lane=row)
// needed by GEMM2. B operands stream as single coalesced b64 fragment loads
// from the repacked (L2-resident) weights.
// ---------------------------------------------------------------------------
__global__ __launch_bounds__(32)
void mlp_wmma(const float* __restrict__ node_feats,
              const float* __restrict__ agg,
              const v2f*  __restrict__ WT1, const float* __restrict__ b1,
              const v2f*  __restrict__ WT2, const float* __restrict__ b2,
              float* __restrict__ out, int N) {
    __shared__ float Xs[16 * XSTR];  // 12544 B
    __shared__ float Hs[16 * HSTR];  // 16640 B

    const int tile = blockIdx.x;
    const size_t r0 = (size_t)tile * 16;
    if (r0 >= (size_t)N) return;

    const int lane = threadIdx.x & 31;
    const int half = lane >> 4;     // 0: K-lo / M rows 0-7, 1: K-hi / M rows 8-15
    const int lr   = lane & 15;     // row for A-frags, column for B/C/D frags
    const int koff = half << 1;     // A K offset within a 4-wide group: 0 or 2

    // ---- Async-stage X = [node_feats | agg] tile (16 x 192 f32) into LDS ----
    for (int idx = lane; idx < 16 * 48; idx += 32) {
        int r = idx / 48;
        int q = idx % 48;           // float4 slot within the 192-wide row
        const float* src = (q < 32)
            ? node_feats + (r0 + r) * NODE_NF + q * 4
            : agg        + (r0 + r) * EDGE_NF + (q - 32) * 4;
        // Generic-pointer low 32 bits == LDS byte address (aperture mapping).
        unsigned dst = (unsigned)(unsigned long long)&Xs[r * XSTR + q * 4];
        asm volatile("global_load_async_to_lds_b128 %0, %1, off"
                     :: "v"(dst), "v"(src) : "memory");
    }
    asm volatile("s_wait_asynccnt 0x0" ::: "memory");
    __syncthreads();

    // ---- GEMM1: h = relu(X @ W1 + b1), 16 column tiles, K = 192 ----
    for (int nt = 0; nt < 16; ++nt) {
        const int n = nt * 16 + lr;
        const float bias = b1[n];
        v8f acc;
        #pragma unroll
        for (int i = 0; i < 8; ++i) acc[i] = bias;

        const v2f* bf = WT1 + (size_t)nt * (IN_NF / 4) * 32 + lane;
        #pragma unroll 8
        for (int kq = 0; kq < IN_NF / 4; ++kq) {
            v2f a = *(const v2f*)(&Xs[lr * XSTR + kq * 4 + koff]);
            v2f b = bf[kq * 32];
            acc = __builtin_amdgcn_wmma_f32_16x16x4_f32(
                false, a, false, b, (short)0, acc, false, false);
        }
        // ReLU + write D tile to LDS (lane holds column n, VGPR i -> row)
        #pragma unroll
        for (int i = 0; i < 8; ++i) {
            int m = half * 8 + i;
            Hs[m * HSTR + n] = fmaxf(acc[i], 0.0f);
        }
    }
    __syncthreads();

    // ---- GEMM2: out = h @ W2 + b2, 8 column tiles, K = 256 ----
    for (int ot = 0; ot < 8; ++ot) {
        const int n = ot * 16 + lr;
        const float bias = b2[n];
        v8f acc;
        #pragma unroll
        for (int i = 0; i < 8; ++i) acc[i] = bias;

        const v2f* bf = WT2 + (size_t)ot * (H_NF / 4) * 32 + lane;
        #pragma unroll 8
        for (int kq = 0; kq < H_NF / 4; ++kq) {
            v2f a = *(const v2f*)(&Hs[lr * HSTR + kq * 4 + koff]);
            v2f b = bf[kq * 32];
            acc = __builtin_amdgcn_wmma_f32_16x16x4_f32(
                false, a, false, b, (short)0, acc, false, false);
        }
        #pragma unroll
        for (int i = 0; i < 8; ++i) {
            int m = half * 8 + i;
            out[(r0 + m) * OUT_NF + n] = acc[i];
        }
    }
}

// ---------------------------------------------------------------------------
extern "C" void kernel_launch(void* const* d_in, const int* in_sizes, int n_in,
                              void* d_out, int out_size, void* d_ws, size_t ws_size,
                              hipStream_t stream) {
    const float* node_feats = (const float*)d_in[0];
    const int*   edge_index = (const int*)d_in[1];   // row = first E entries
    const float* edge_attr  = (const float*)d_in[2];
    const float* W1 = (const float*)d_in[3];
    const float* b1 = (const float*)d_in[4];
    const float* W2 = (const float*)d_in[5];
    const float* b2 = (const float*)d_in[6];
    float* out = (float*)d_out;

    const int N = in_sizes[0] / NODE_NF;             // 50000
    const int E = in_sizes[2] / EDGE_NF;             // 800000

    // Workspace layout: agg (N*64 f32) | WT1 frags | WT2 frags
    float* agg = (float*)d_ws;
    size_t aggBytes = ((size_t)N * EDGE_NF * sizeof(float) + 255) & ~(size_t)255;
    v2f* WT1 = (v2f*)((char*)d_ws + aggBytes);       // 24576 v2f = 196608 B
    v2f* WT2 = (v2f*)((char*)d_ws + aggBytes + (size_t)(IN_NF/4)*(H_NF/16)*32*8);

    // 0) repack weights into B-fragment order (idempotent, tiny)
    {
        int t1 = (IN_NF / 4) * (H_NF / 16) * 32;     // 24576
        int t2 = (H_NF / 4) * (OUT_NF / 16) * 32;    // 16384
        make_frags<<<(t1 + 255) / 256, 256, 0, stream>>>(W1, WT1, IN_NF, H_NF);
        make_frags<<<(t2 + 255) / 256, 256, 0, stream>>>(W2, WT2, H_NF, OUT_NF);
    }
    // 1) zero agg
    {
        int n4 = (N * EDGE_NF) / 4;
        zero_agg<<<(n4 + 255) / 256, 256, 0, stream>>>(agg, n4);
    }
    // 2) scatter-add edges
    {
        long long threads = (long long)E * 16;
        scatter_edges<<<(unsigned)((threads + 255) / 256), 256, 0, stream>>>(
            edge_attr, edge_index, agg, E);
    }
    // 3) fused concat + MLP (one wave per 16-row tile)
    {
        int tiles = (N + 15) / 16;                   // 3125
        mlp_wmma<<<tiles, 32, 0, stream>>>(node_feats, agg, WT1, b1, WT2, b2, out, N);
    }
}